// GradientRefinementModule_34634616275301
// MI455X (gfx1250) — compile-verified
//
#include <hip/hip_runtime.h>

// GradientRefinementModule: 65536 independent rows, 3 iterations of
// gradient refinement on positions [:,0] and [:,2] against a 2048-float
// signal row, linear-interp sampling, final order/swap.
// Memory-latency-bound gather workload: one thread per row.

#define SIGNAL_LENGTH 2048
#define NUM_ITERATIONS 3
#define STEP_SIZE 0.001f
// 10.0 / 4.0 / 2048 == 0.001220703125 (exact in fp32)
#define DERIV_STEP 0.001220703125f

__device__ __forceinline__ float clamp01(float x) {
    return fminf(fmaxf(x, 0.0f), 1.0f);
}

__device__ __forceinline__ int clampi(int x, int lo, int hi) {
    return x < lo ? lo : (x > hi ? hi : x);
}

// Linear-interp sample, literal transcription of reference _sample.
__device__ __forceinline__ float sample1(const float* __restrict__ s, float pos) {
    float ind = pos * (float)(SIGNAL_LENGTH - 1);
    int il = clampi((int)floorf(ind), 0, SIGNAL_LENGTH - 1);
    int ir = clampi((int)ceilf(ind),  0, SIGNAL_LENGTH - 1);
    float wr = ind - (float)il;
    float wl = 1.0f - wr;
    float vl = s[il];
    float vr = s[ir];
    return vl * wl + vr * wr;
}

// grad + curvature via central differences (reference _grad_curv).
__device__ __forceinline__ void grad_curv(const float* __restrict__ s, float pos,
                                          float& g, float& c) {
    float pl = clamp01(pos - DERIV_STEP);
    float pr = clamp01(pos + DERIV_STEP);
    float v  = sample1(s, pos);
    float vl = sample1(s, pl);
    float vr = sample1(s, pr);
    g = (vr - vl) / (2.0f * DERIV_STEP);
    c = (vr - 2.0f * v + vl) / (DERIV_STEP * DERIV_STEP);
}

__global__ void __launch_bounds__(256)
GradientRefinementModule_kernel(const float* __restrict__ signal,
                                const float* __restrict__ peak_positions,
                                float* __restrict__ out,
                                int n_rows) {
    int row = blockIdx.x * blockDim.x + threadIdx.x;
    if (row >= n_rows) return;

    const float* __restrict__ s = signal + (size_t)row * SIGNAL_LENGTH;

    float p0 = peak_positions[row * 3 + 0];
    float p2 = peak_positions[row * 3 + 2];

    // Kick the two hot cachelines into the hierarchy early
    // (lowers to global_prefetch_b8 on gfx1250). Iteration 1's 12 gathers
    // all fall inside these windows.
    {
        int i0 = clampi((int)(p0 * (float)(SIGNAL_LENGTH - 1)), 0, SIGNAL_LENGTH - 1);
        int i2 = clampi((int)(p2 * (float)(SIGNAL_LENGTH - 1)), 0, SIGNAL_LENGTH - 1);
        __builtin_prefetch(s + i0, 0, 3);
        __builtin_prefetch(s + i2, 0, 3);
    }

#pragma unroll
    for (int it = 0; it < NUM_ITERATIONS; ++it) {
        float g1, c1, g2, c2;
        grad_curv(s, p0, g1, c1);
        grad_curv(s, p2, g2, c2);
        float a1 = (c1 < 0.0f) ? (-STEP_SIZE * g1) : 0.0f;
        float a2 = (c2 < 0.0f) ? (-STEP_SIZE * g2) : 0.0f;
        p0 = clamp01(p0 + a1);
        p2 = clamp01(p2 + a2);
        // mid is computed once from the final (p0,p2) after the loop and never feeds
        // back into the refinement — only the final value matters.
    }

    float mid = 0.5f * (p0 + p2);
    bool ordered = p0 < p2;
    float o0 = ordered ? p0 : p2;
    float o2 = ordered ? p2 : p0;

    out[row * 3 + 0] = o0;
    out[row * 3 + 1] = mid;
    out[row * 3 + 2] = o2;
}

extern "C" void kernel_launch(void* const* d_in, const int* in_sizes, int n_in,
                              void* d_out, int out_size, void* d_ws, size_t ws_size,
                              hipStream_t stream) {
    const float* signal         = (const float*)d_in[0];
    const float* peak_positions = (const float*)d_in[1];
    float* out                  = (float*)d_out;

    int n_rows = in_sizes[1] / 3;   // 65536

    const int block = 256;          // 8 wave32s per block
    const int grid  = (n_rows + block - 1) / block;
    GradientRefinementModule_kernel<<<grid, block, 0, stream>>>(
        signal, peak_positions, out, n_rows);
}